// MM_final_77180562309363
// MI455X (gfx1250) — compile-verified
//
#include <hip/hip_runtime.h>
#include <hip/hip_bf16.h>

typedef float v2f __attribute__((ext_vector_type(2)));
typedef float v8f __attribute__((ext_vector_type(8)));

#define C_   64
#define D_   8
#define K15_ 15
#define K16_ 16
#define OUT_ 512

// ---------------------------------------------------------------------------
// Kernel 1: per (16-row b-tile, channel c) compute idx[b,c] = argmax_j h[b,c,j]
// using two chained WMMA f32 16x16x4 GEMMs.
//   GEMM1: P(16x16) = X_tile(16x8) * S_c(8x16 zero-padded), minus T, sign
//   GEMM2: h(16x16) = s(16x16, col15=0) * Hpad(16x16, row15=0)
// One wave per channel; 8 waves (8 channels) per block.
// Zero-padding is done branchlessly: clamped-index load + v_cndmask select.
// ---------------------------------------------------------------------------
__global__ __launch_bounds__(256) void kn_idx(const float* __restrict__ x,
                                              const float* __restrict__ S,
                                              const float* __restrict__ H,
                                              const float* __restrict__ T,
                                              unsigned char* __restrict__ idx) {
  __shared__ float tile[8][16 * 17];   // per-wave 16x16 scratch, padded stride

  const int lane = threadIdx.x & 31;
  const int wave = threadIdx.x >> 5;
  const int b0   = blockIdx.x * 16;
  const int c    = blockIdx.y * 8 + wave;

  float* sm = &tile[wave][0];

  const int  n      = lane & 15;        // N column for B/C/D layouts
  const int  m      = n;                // M row for A layout (lane%16)
  const int  hiHalf = lane >> 4;        // 0: lanes 0-15, 1: lanes 16-31
  const int  nc     = (n < K15_) ? n : (K15_ - 1);  // clamped column
  const bool nOK    = (n < K15_);

  // ---------------- GEMM1: P = X * S_c  (K=8 -> two K=4 chunks) ------------
  v8f p = {};
#pragma unroll
  for (int q = 0; q < 2; ++q) {
    const int kb = 4 * q + 2 * hiHalf;           // K base for this lane half
    v2f a, b;
    // A layout: lane m holds (m, kb) and (m, kb+1) — unconditional loads
    a.x = x[(b0 + m) * (C_ * D_) + c * D_ + kb];
    a.y = x[(b0 + m) * (C_ * D_) + c * D_ + kb + 1];
    // B layout: lane n holds (kb, n), (kb+1, n); branchless n==15 padding
    float s0 = S[(c * D_ + kb) * K15_ + nc];
    float s1 = S[(c * D_ + kb + 1) * K15_ + nc];
    b.x = nOK ? s0 : 0.0f;
    b.y = nOK ? s1 : 0.0f;
    p = __builtin_amdgcn_wmma_f32_16x16x4_f32(false, a, false, b,
                                              (short)0, p, false, false);
  }

  // ---------------- subtract T, take sign, relayout C/D -> A via LDS -------
  const float tload = T[c * K15_ + nc];
  const float tv    = nOK ? tload : 0.0f;
#pragma unroll
  for (int v = 0; v < 8; ++v) {
    const float pv = p[v] - tv;
    const float sv = nOK ? (pv > 0.0f ? 1.0f : -1.0f) : 0.0f;
    sm[(v + 8 * hiHalf) * 17 + n] = sv;          // s[row][k]
  }
  __syncthreads();

  // ---------------- GEMM2: h = s * Hpad (K=16 -> four K=4 chunks) ----------
  v8f h = {};
#pragma unroll
  for (int q = 0; q < 4; ++q) {
    const int kb  = 4 * q + 2 * hiHalf;
    const int r0  = (kb     < K15_) ? kb       : (K15_ - 1);  // clamped rows
    const int r1  = (kb + 1 < K15_) ? (kb + 1) : (K15_ - 1);
    v2f a, b;
    a.x = sm[m * 17 + kb];
    a.y = sm[m * 17 + kb + 1];
    float h0 = H[r0 * K16_ + n];
    float h1 = H[r1 * K16_ + n];
    b.x = (kb     < K15_) ? h0 : 0.0f;
    b.y = (kb + 1 < K15_) ? h1 : 0.0f;
    h = __builtin_amdgcn_wmma_f32_16x16x4_f32(false, a, false, b,
                                              (short)0, h, false, false);
  }
  __syncthreads();

  // ---------------- argmax over j per b-row --------------------------------
#pragma unroll
  for (int v = 0; v < 8; ++v)
    sm[(v + 8 * hiHalf) * 17 + n] = h[v];        // h[row][j]
  __syncthreads();

  if (lane < 16) {
    const int row = lane;
    float best = sm[row * 17 + 0];
    int   bi   = 0;
#pragma unroll
    for (int j = 1; j < 16; ++j) {
      const float vv = sm[row * 17 + j];
      if (vv > best) { best = vv; bi = j; }      // first-max wins (matches argmax)
    }
    idx[(b0 + row) * C_ + c] = (unsigned char)bi;
  }
}

// ---------------------------------------------------------------------------
// Kernel 2: out[b, o] = sum_c LUT[c, idx[b,c], o]
// Each block owns a 32-wide out-column tile; the full 64x16x32 LUT slice
// (128 KB) lives in dynamic LDS for the block's whole row range.
// Thread t: o_local = t%32 (one wave == one b-row, coalesced), r = t/32.
// ---------------------------------------------------------------------------
__global__ __launch_bounds__(256) void kn_acc(const float* __restrict__ LUT,
                                              const unsigned char* __restrict__ idx,
                                              float* __restrict__ out,
                                              int rowsPerBlock) {
  extern __shared__ float lutS[];                        // 64*16*32 floats
  unsigned char* idxS = (unsigned char*)(lutS + C_ * K16_ * 32); // 8*64 bytes

  const int tid = threadIdx.x;
  const int o0  = blockIdx.y * 32;

  // cooperative load of LUT[:, :, o0:o0+32] into LDS (float4 vectors)
  for (int i = tid; i < (C_ * K16_ * 32) / 4; i += 256) {
    const int e  = i * 4;
    const int ck = e >> 5;     // (c*16 + k)
    const int o  = e & 31;
    *(float4*)(lutS + e) = *(const float4*)(LUT + ck * OUT_ + o0 + o);
  }

  const int oL     = tid & 31;
  const int r      = tid >> 5;                 // 0..7 : b-row within group
  const int bStart = blockIdx.x * rowsPerBlock;

  for (int bb = 0; bb < rowsPerBlock; bb += 8) {
    __syncthreads();                           // lutS ready / idxS reusable
    if (tid < 128)                             // stage 8 rows of idx (512 B)
      ((unsigned int*)idxS)[tid] =
          ((const unsigned int*)(idx + (bStart + bb) * C_))[tid];
    __syncthreads();

    const unsigned char* ir = idxS + r * C_;
    float acc = 0.0f;
#pragma unroll 8
    for (int cc = 0; cc < C_; ++cc) {
      const int k = ir[cc];                    // LDS broadcast within wave
      acc += lutS[(cc * K16_ + k) * 32 + oL];  // conflict-free: 32 consec words
    }
    out[(long long)(bStart + bb + r) * OUT_ + o0 + oL] = acc;
  }
}

// ---------------------------------------------------------------------------
extern "C" void kernel_launch(void* const* d_in, const int* in_sizes, int n_in,
                              void* d_out, int out_size, void* d_ws, size_t ws_size,
                              hipStream_t stream) {
  const float* x   = (const float*)d_in[0];
  const float* S   = (const float*)d_in[1];
  const float* H   = (const float*)d_in[2];
  const float* T   = (const float*)d_in[3];
  const float* LUT = (const float*)d_in[4];
  float* out = (float*)d_out;

  unsigned char* idx = (unsigned char*)d_ws;   // needs B*C = 2 MB of scratch
  const int Btot = in_sizes[0] / (C_ * D_);    // 32768

  dim3 g1(Btot / 16, C_ / 8);                  // 2048 x 8 blocks, 8 waves each
  kn_idx<<<g1, 256, 0, stream>>>(x, S, H, T, idx);

  const int bsplit = 64;                       // rows split across blocks
  dim3 g2(bsplit, OUT_ / 32);                  // 64 x 16 = 1024 blocks
  const size_t dynLds = C_ * K16_ * 32 * sizeof(float) + 8 * C_; // 128 KB + 512 B
  kn_acc<<<g2, 256, dynLds, stream>>>(LUT, idx, out, Btot / bsplit);
}